// GNN_33844342293302
// MI455X (gfx1250) — compile-verified
//
#include <hip/hip_runtime.h>
#include <hip/hip_bf16.h>

typedef float v2f __attribute__((ext_vector_type(2)));
typedef float v8f __attribute__((ext_vector_type(8)));

#define WAVES_PER_BLOCK 8

// ---------------- scratch zeroing (must run every call) ----------------
__global__ __launch_bounds__(256) void gnn_zero(float* __restrict__ p, size_t n) {
    size_t i = (size_t)blockIdx.x * blockDim.x + threadIdx.x;
    size_t stride = (size_t)gridDim.x * blockDim.x;
    for (; i < n; i += stride) p[i] = 0.0f;
}

// ---------------- pad x[N][6] -> xp[N][8] (cols 6,7 = 0), uniform control flow ----------------
__global__ __launch_bounds__(256) void gnn_padx(const float* __restrict__ x,
                                                float* __restrict__ xp, int N) {
    int i = blockIdx.x * blockDim.x + threadIdx.x;
    if (i >= N) return;
    const float* xs = x + (size_t)i * 6;
    float4 a = {xs[0], xs[1], xs[2], xs[3]};
    float4 b = {xs[4], xs[5], 0.0f, 0.0f};
    float4* o = (float4*)(xp + (size_t)i * 8);
    o[0] = a;
    o[1] = b;
}

// ---------------- pack padded weights into ws (single block) ----------------
// wpad layout: W1lp[8][16] | W1rp[8][16] | Wcp[16][16] | bcp[16]   (528 floats)
__global__ __launch_bounds__(256) void gnn_pack(const float* __restrict__ W1l,
                                                const float* __restrict__ W1r,
                                                const float* __restrict__ Wc,
                                                const float* __restrict__ bc,
                                                float* __restrict__ wpad) {
    int t = threadIdx.x;
    int k = t >> 4, c = t & 15;
    if (t < 128) {
        wpad[t]       = (k < 6) ? W1l[k * 16 + c] : 0.0f;
        wpad[128 + t] = (k < 6) ? W1r[k * 16 + c] : 0.0f;
    }
    wpad[256 + t] = (c < 8) ? Wc[k * 8 + c] : 0.0f;   // 16x16, cols 8..15 zero
    if (t < 16) wpad[512 + t] = (t < 8) ? bc[t] : 0.0f;
}

// ---------------- edge scatter, layer 1 (6 feats) + degree ----------------
__global__ __launch_bounds__(256) void gnn_scatter1(
    const long long* __restrict__ src, const long long* __restrict__ dst,
    const float* __restrict__ x, float* __restrict__ agg1,
    float* __restrict__ deg, int E) {
    int e = blockIdx.x * blockDim.x + threadIdx.x;
    if (e >= E) return;
    long long s = src[e];
    long long d = dst[e];
    const float* xs = x + s * 6;
    float* a = agg1 + d * 8;          // stride-8 rows, cols 0..5 used, 6..7 stay zero
    atomicAdd(&deg[d], 1.0f);
#pragma unroll
    for (int f = 0; f < 6; ++f) atomicAdd(&a[f], xs[f]);
}

// ---------------- inv_deg = 1/max(deg,1), in place ----------------
__global__ __launch_bounds__(256) void gnn_inv(float* __restrict__ deg, int n) {
    int i = blockIdx.x * blockDim.x + threadIdx.x;
    if (i < n) deg[i] = 1.0f / fmaxf(deg[i], 1.0f);
}

// ---------------- layer 1: h1 = relu(mean@W1l + x@W1r + b1), WMMA f32 ----------------
__global__ __launch_bounds__(256) void gnn_layer1(
    const float* __restrict__ xp, const float* __restrict__ agg1,
    const float* __restrict__ inv_deg,
    const float* __restrict__ W1lp, const float* __restrict__ W1rp,
    const float* __restrict__ b1, float* __restrict__ h1,
    int n_tiles, int N) {
    int wave = threadIdx.x >> 5;
    int lane = threadIdx.x & 31;
    int tile = blockIdx.x * WAVES_PER_BLOCK + wave;
    if (tile >= n_tiles) return;

    int base = tile * 16;
    int col  = lane & 15;        // B/C/D column; also A row-in-tile
    int hi   = lane >> 4;
    int kb   = hi * 2;
    int row  = base + col;
    int rowc = (row < N) ? row : (N - 1);   // clamp: masked at store (tail tile only)
    float inv = inv_deg[rowc];

    v8f c;
    float bias = b1[col];
#pragma unroll
    for (int i = 0; i < 8; ++i) c[i] = bias;

#pragma unroll
    for (int chunk = 0; chunk < 2; ++chunk) {
        int k0 = chunk * 4 + kb;                       // even -> 8B aligned
        v2f am = *(const v2f*)(agg1 + (size_t)rowc * 8 + k0);
        v2f ax = *(const v2f*)(xp   + (size_t)rowc * 8 + k0);
        am *= inv;
        v2f bl, br;
        bl.x = W1lp[k0 * 16 + col];
        bl.y = W1lp[(k0 + 1) * 16 + col];
        br.x = W1rp[k0 * 16 + col];
        br.y = W1rp[(k0 + 1) * 16 + col];
        c = __builtin_amdgcn_wmma_f32_16x16x4_f32(false, am, false, bl, (short)0, c, false, false);
        c = __builtin_amdgcn_wmma_f32_16x16x4_f32(false, ax, false, br, (short)0, c, false, false);
    }

    // fast path: full tile (wave-uniform branch, no per-row exec churn)
    float* hrow = h1 + (size_t)(base + hi * 8) * 16 + col;
    if (base + 16 <= N) {
#pragma unroll
        for (int r = 0; r < 8; ++r) hrow[(size_t)r * 16] = fmaxf(c[r], 0.0f);
    } else {
#pragma unroll
        for (int r = 0; r < 8; ++r) {
            int n = base + r + hi * 8;
            if (n < N) h1[(size_t)n * 16 + col] = fmaxf(c[r], 0.0f);
        }
    }
}

// ---------------- edge scatter, layer 2 (16 feats) ----------------
__global__ __launch_bounds__(256) void gnn_scatter2(
    const long long* __restrict__ src, const long long* __restrict__ dst,
    const float* __restrict__ h1, float* __restrict__ agg2, int E) {
    int e = blockIdx.x * blockDim.x + threadIdx.x;
    if (e >= E) return;
    long long s = src[e];
    long long d = dst[e];
    const float4* hs = (const float4*)(h1 + s * 16);
    float4 v0 = hs[0], v1 = hs[1], v2 = hs[2], v3 = hs[3];
    float* a = agg2 + d * 16;
    atomicAdd(&a[0],  v0.x); atomicAdd(&a[1],  v0.y);
    atomicAdd(&a[2],  v0.z); atomicAdd(&a[3],  v0.w);
    atomicAdd(&a[4],  v1.x); atomicAdd(&a[5],  v1.y);
    atomicAdd(&a[6],  v1.z); atomicAdd(&a[7],  v1.w);
    atomicAdd(&a[8],  v2.x); atomicAdd(&a[9],  v2.y);
    atomicAdd(&a[10], v2.z); atomicAdd(&a[11], v2.w);
    atomicAdd(&a[12], v3.x); atomicAdd(&a[13], v3.y);
    atomicAdd(&a[14], v3.z); atomicAdd(&a[15], v3.w);
}

// ---------------- layer 2 + classifier, fused via per-wave LDS tile ----------------
__global__ __launch_bounds__(256) void gnn_layer2_cls(
    const float* __restrict__ h1, const float* __restrict__ agg2,
    const float* __restrict__ inv_deg,
    const float* __restrict__ W2l, const float* __restrict__ W2r,
    const float* __restrict__ b2,
    const float* __restrict__ Wcp, const float* __restrict__ bcp,
    float* __restrict__ out, int n_tiles, int N) {
    __shared__ float tile_lds[WAVES_PER_BLOCK][16][17];  // +1 pad: conflict-free column reads

    int wave = threadIdx.x >> 5;
    int lane = threadIdx.x & 31;
    int tile = blockIdx.x * WAVES_PER_BLOCK + wave;
    if (tile >= n_tiles) return;

    int base = tile * 16;
    int col  = lane & 15;
    int hi   = lane >> 4;
    int kb   = hi * 2;
    int row  = base + col;
    int rowc = (row < N) ? row : (N - 1);
    float inv = inv_deg[rowc];

    v8f c;
    float bias = b2[col];
#pragma unroll
    for (int i = 0; i < 8; ++i) c[i] = bias;

#pragma unroll
    for (int chunk = 0; chunk < 4; ++chunk) {
        int k0 = chunk * 4 + kb;                       // even -> 8B aligned
        v2f am = *(const v2f*)(agg2 + (size_t)rowc * 16 + k0);
        v2f ah = *(const v2f*)(h1   + (size_t)rowc * 16 + k0);
        am *= inv;
        v2f bl, br;
        bl.x = W2l[k0 * 16 + col];
        bl.y = W2l[(k0 + 1) * 16 + col];
        br.x = W2r[k0 * 16 + col];
        br.y = W2r[(k0 + 1) * 16 + col];
        c = __builtin_amdgcn_wmma_f32_16x16x4_f32(false, am, false, bl, (short)0, c, false, false);
        c = __builtin_amdgcn_wmma_f32_16x16x4_f32(false, ah, false, br, (short)0, c, false, false);
    }

    // ReLU(h2) -> per-wave LDS tile (re-layout D(M striped over halves) -> A(row per lane))
    float (*t)[17] = tile_lds[wave];
#pragma unroll
    for (int r = 0; r < 8; ++r) {
        int m = r + hi * 8;
        t[m][col] = fmaxf(c[r], 0.0f);
    }
    asm volatile("s_wait_dscnt 0" ::: "memory");  // same-wave LDS RAW ordering

    // classifier: out = h2 @ Wcp + bcp  (Wcp pre-padded 16x16, cols 8..15 zero)
    v8f cc;
    float cbias = bcp[col];
#pragma unroll
    for (int i = 0; i < 8; ++i) cc[i] = cbias;

#pragma unroll
    for (int chunk = 0; chunk < 4; ++chunk) {
        int k0 = chunk * 4 + kb;
        v2f a, b;
        a.x = t[col][k0];
        a.y = t[col][k0 + 1];
        b.x = Wcp[k0 * 16 + col];
        b.y = Wcp[(k0 + 1) * 16 + col];
        cc = __builtin_amdgcn_wmma_f32_16x16x4_f32(false, a, false, b, (short)0, cc, false, false);
    }

    // fast path: full tile -> single lane-mask, 8 unconditional stores
    if (base + 16 <= N) {
        if (col < 8) {
            float* orow = out + (size_t)(base + hi * 8) * 8 + col;
#pragma unroll
            for (int r = 0; r < 8; ++r) orow[(size_t)r * 8] = cc[r];
        }
    } else {
        if (col < 8) {
#pragma unroll
            for (int r = 0; r < 8; ++r) {
                int n = base + r + hi * 8;
                if (n < N) out[(size_t)n * 8 + col] = cc[r];
            }
        }
    }
}

extern "C" void kernel_launch(void* const* d_in, const int* in_sizes, int n_in,
                              void* d_out, int out_size, void* d_ws, size_t ws_size,
                              hipStream_t stream) {
    const float*     x   = (const float*)d_in[0];
    const long long* ei  = (const long long*)d_in[1];   // int64 edge_index [2, E]
    const float*     W1l = (const float*)d_in[2];
    const float*     W1r = (const float*)d_in[3];
    const float*     b1  = (const float*)d_in[4];
    const float*     W2l = (const float*)d_in[5];
    const float*     W2r = (const float*)d_in[6];
    const float*     b2  = (const float*)d_in[7];
    const float*     Wc  = (const float*)d_in[8];
    const float*     bc  = (const float*)d_in[9];
    float*           out = (float*)d_out;

    const int N = in_sizes[0] / 6;
    const int E = in_sizes[1] / 2;

    // ws layout (floats):
    //   deg[N] | agg1[8N] | agg2[16N] | h1[16N] | xp[8N] | wpad[528]
    // first 25N floats (deg+agg1+agg2) must be zeroed every call.
    float* ws   = (float*)d_ws;
    float* deg  = ws;
    float* agg1 = ws + (size_t)N;
    float* agg2 = ws + (size_t)N * 9;
    float* h1   = ws + (size_t)N * 25;
    float* xp   = ws + (size_t)N * 41;
    float* wpad = ws + (size_t)N * 49;
    const size_t zero_n = (size_t)N * 25;

    float* W1lp = wpad;
    float* W1rp = wpad + 128;
    float* Wcp  = wpad + 256;
    float* bcp  = wpad + 512;

    const long long* src = ei;
    const long long* dst = ei + E;

    const int TB = 256;
    int zgrid = (int)((zero_n + TB - 1) / TB);
    if (zgrid > 16384) zgrid = 16384;
    gnn_zero<<<zgrid, TB, 0, stream>>>(ws, zero_n);

    gnn_padx<<<(N + TB - 1) / TB, TB, 0, stream>>>(x, xp, N);
    gnn_pack<<<1, TB, 0, stream>>>(W1l, W1r, Wc, bc, wpad);

    gnn_scatter1<<<(E + TB - 1) / TB, TB, 0, stream>>>(src, dst, x, agg1, deg, E);
    gnn_inv<<<(N + TB - 1) / TB, TB, 0, stream>>>(deg, N);

    const int n_tiles = (N + 15) / 16;
    const int tgrid = (n_tiles + WAVES_PER_BLOCK - 1) / WAVES_PER_BLOCK;
    gnn_layer1<<<tgrid, TB, 0, stream>>>(xp, agg1, deg, W1lp, W1rp, b1, h1, n_tiles, N);

    gnn_scatter2<<<(E + TB - 1) / TB, TB, 0, stream>>>(src, dst, h1, agg2, E);

    gnn_layer2_cls<<<tgrid, TB, 0, stream>>>(h1, agg2, deg, W2l, W2r, b2, Wcp, bcp,
                                             out, n_tiles, N);
}